// SimpleGCN_4200478016008
// MI455X (gfx1250) — compile-verified
//
#include <hip/hip_runtime.h>
#include <hip/hip_bf16.h>

// GCN (2-layer, PyG GCNConv semantics) for MI455X / gfx1250.
// Pipeline (all stream-ordered):
//   1) deg[i]=1 (self-loop)           4) gemm1: xws=(x@W1)*dinv -> bufA, and acc1 init -> bufB
//   2) deg[col[e]] += 1 (atomics)     5) scatter1: acc1[col] += xws[row]
//   3) dinv = rsqrt(deg)              6) h = relu(dinv*acc1 + b1) -> bufA
//                                     7) gemm2: hws=(h@W2)*dinv -> bufB, acc2 init -> d_out
//                                     8) scatter2: d_out[col] += hws[row]
//                                     9) d_out = dinv*d_out + b2  (in place)

typedef __attribute__((ext_vector_type(2))) float v2f;
typedef __attribute__((ext_vector_type(8))) float v8f;

// ---------------- elementwise helpers ----------------

__global__ void k_fill1(float* __restrict__ p, int n) {
    int i = blockIdx.x * blockDim.x + threadIdx.x;
    if (i < n) p[i] = 1.0f;
}

__global__ void k_deg(const int* __restrict__ col, float* __restrict__ deg, int E) {
    int e = blockIdx.x * blockDim.x + threadIdx.x;
    if (e < E) atomicAdd(&deg[col[e]], 1.0f);
}

__global__ void k_dinv(const float* __restrict__ deg, float* __restrict__ dinv, int n) {
    int i = blockIdx.x * blockDim.x + threadIdx.x;
    if (i < n) dinv[i] = rsqrtf(deg[i]);   // deg >= 1 always (self-loop)
}

// h = relu(dinv[i]*acc[i,f] + b[f]);  dim = 64
__global__ void k_relu_bias(const float* __restrict__ acc, const float* __restrict__ dinv,
                            const float* __restrict__ b, float* __restrict__ h, int total) {
    int g = blockIdx.x * blockDim.x + threadIdx.x;
    if (g >= total) return;
    int i = g >> 6, f = g & 63;
    float v = dinv[i] * acc[g] + b[f];
    h[g] = v > 0.0f ? v : 0.0f;
}

// out = dinv[i]*out + b[f] (in place); dim = 32
__global__ void k_final(float* __restrict__ out, const float* __restrict__ dinv,
                        const float* __restrict__ b, int total) {
    int g = blockIdx.x * blockDim.x + threadIdx.x;
    if (g >= total) return;
    int i = g >> 5, f = g & 31;
    out[g] = dinv[i] * out[g] + b[f];
}

// ---------------- WMMA GEMM: outS = (A[M,64] @ W[64,NC]) * dinv[row] ----------------
// One wave computes a 16-row x NC block using V_WMMA_F32_16X16X4_F32 (full fp32).
// Also duplicates the result into outAcc (self-loop init of the scatter accumulator).

template <int NC>
__global__ void gcn_gemm_wmma(const float* __restrict__ A, const float* __restrict__ W,
                              const float* __restrict__ dinv,
                              float* __restrict__ outS, float* __restrict__ outAcc,
                              int Mrows) {
    const int lane = threadIdx.x & 31;
    const int wave = threadIdx.x >> 5;
    const int tile = blockIdx.x * (blockDim.x >> 5) + wave;
    const int row0 = tile * 16;
    if (row0 >= Mrows) return;               // wave-uniform: EXEC all-ones at WMMA

    const int m    = lane & 15;              // M index (A) / N index (B,C)
    const int half = lane >> 4;              // half-wave select
    constexpr int NT = NC / 16;

    v8f acc[NT];
#pragma unroll
    for (int j = 0; j < NT; ++j) acc[j] = (v8f){};

    // A: lane holds A[row0+m][k0 + 2*half + {0,1}]
    const float* arow = A + (size_t)(row0 + m) * 64 + 2 * half;

#pragma unroll
    for (int k0 = 0; k0 < 64; k0 += 4) {
        v2f a = *reinterpret_cast<const v2f*>(arow + k0);
#pragma unroll
        for (int j = 0; j < NT; ++j) {
            // B (4x16): VGPR i -> row K = k0 + i + 2*half, col N = j*16 + m
            const float* wp = W + (size_t)(k0 + 2 * half) * NC + j * 16 + m;
            v2f b;
            b.x = wp[0];
            b.y = wp[NC];
            acc[j] = __builtin_amdgcn_wmma_f32_16x16x4_f32(
                false, a, false, b, (short)0, acc[j], false, false);
        }
    }

    // C/D layout: VGPR v, lanes0-15 -> M=v, lanes16-31 -> M=v+8; N = m
    float s[8];
#pragma unroll
    for (int v = 0; v < 8; ++v) s[v] = dinv[row0 + v + 8 * half];

#pragma unroll
    for (int j = 0; j < NT; ++j) {
#pragma unroll
        for (int v = 0; v < 8; ++v) {
            int r = row0 + v + 8 * half;
            float val = acc[j][v] * s[v];
            size_t o = (size_t)r * NC + j * 16 + m;
            outS[o]   = val;   // scaled features (gather source)
            outAcc[o] = val;   // accumulator init == self-loop contribution
        }
    }
}

// ---------------- scatter: dst[col[e], :] += src[row[e], :] ----------------
// One wave per edge; lanes stride the feature dim (coalesced gather, f32 atomics to L2).

__global__ void gcn_scatter(const int* __restrict__ rows, const int* __restrict__ cols,
                            const float* __restrict__ src, float* __restrict__ dst,
                            int E, int dim) {
    const int lane = threadIdx.x & 31;
    const int e = blockIdx.x * (blockDim.x >> 5) + (threadIdx.x >> 5);
    if (e >= E) return;
    const int r = rows[e];
    const int c = cols[e];
    const float* sp = src + (size_t)r * dim;
    float* dp = dst + (size_t)c * dim;
    for (int f = lane; f < dim; f += 32)
        atomicAdd(dp + f, sp[f]);
}

// ---------------- launcher ----------------

extern "C" void kernel_launch(void* const* d_in, const int* in_sizes, int n_in,
                              void* d_out, int out_size, void* d_ws, size_t ws_size,
                              hipStream_t stream) {
    const float* x   = (const float*)d_in[0];
    const int*   ei  = (const int*)d_in[1];   // edge_index flat [2,E]; harness: integer -> int32
    const float* W1  = (const float*)d_in[2];
    const float* b1  = (const float*)d_in[3];
    const float* W2  = (const float*)d_in[4];
    const float* b2  = (const float*)d_in[5];

    const int N = in_sizes[0] / 64;     // 100000
    const int E = in_sizes[1] / 2;      // 1600000
    const int* rows = ei;
    const int* cols = ei + E;

    float* ws   = (float*)d_ws;
    float* deg  = ws;                            // [N]
    float* dinv = ws + N;                        // [N]
    float* bufA = ws + 2 * (size_t)N;            // [N,64]  xws, then h
    float* bufB = bufA + (size_t)N * 64;         // [N,64]  acc1, then hws[N,32]
    float* out  = (float*)d_out;                 // [N,32]  also acc2

    const int T = 256;
    const int wavesPerBlk = T / 32;

    // 1) deg = 1 (self-loops)
    k_fill1<<<(N + T - 1) / T, T, 0, stream>>>(deg, N);
    // 2) deg[col] += 1
    k_deg<<<(E + T - 1) / T, T, 0, stream>>>(cols, deg, E);
    // 3) dinv = rsqrt(deg)
    k_dinv<<<(N + T - 1) / T, T, 0, stream>>>(deg, dinv, N);

    // 4) xws = (x @ W1) * dinv  -> bufA; acc1 init -> bufB
    const int tiles = (N + 15) / 16;
    const int gemmBlks = (tiles + wavesPerBlk - 1) / wavesPerBlk;
    gcn_gemm_wmma<64><<<gemmBlks, T, 0, stream>>>(x, W1, dinv, bufA, bufB, N);

    // 5) acc1[col] += xws[row]
    gcn_scatter<<<(E + wavesPerBlk - 1) / wavesPerBlk, T, 0, stream>>>(rows, cols, bufA, bufB, E, 64);

    // 6) h = relu(dinv*acc1 + b1) -> bufA
    k_relu_bias<<<((N * 64) + T - 1) / T, T, 0, stream>>>(bufB, dinv, b1, bufA, N * 64);

    // 7) hws = (h @ W2) * dinv -> bufB; acc2 init -> d_out
    gcn_gemm_wmma<32><<<gemmBlks, T, 0, stream>>>(bufA, W2, dinv, bufB, out, N);

    // 8) acc2[col] += hws[row]
    gcn_scatter<<<(E + wavesPerBlk - 1) / wavesPerBlk, T, 0, stream>>>(rows, cols, bufB, out, E, 32);

    // 9) out = dinv*out + b2
    k_final<<<((N * 32) + T - 1) / T, T, 0, stream>>>(out, dinv, b2, N * 32);
}